// CFGMentionProposer_39479339385014
// MI455X (gfx1250) — compile-verified
//
#include <hip/hip_runtime.h>
#include <hip/hip_bf16.h>

// CKY inside-outside (log semiring), B x 128 x 128 charts.
// One workgroup (8 wave32) per batch element; chart lives in LDS.
//   Forward: Valiant-blocked inside algorithm.
//     - 16x16 tiles, 8 block diagonals.
//     - Strictly-interior split blocks -> log-matmul-exp on 16x16x16 tiles
//       via 4x v_wmma_f32_16x16x4_f32 per tile (one block pair per wave);
//       cross-lane row-max gather via ds_bpermute_b32; WMMA operands built
//       from direct (divergent-address) LDS loads, no register-indexed arrays.
//     - Boundary splits (<=31 per cell) -> in-wave width-ordered sweep.
//   Backward: inside-outside gradient, softmax normalizer taken analytically
//     (Q = e^{-c_ij} = sigmoid(-s_ij), precomputed into out_marg as scratch),
//     scatter with LDS float atomics.
//   Data movement: per-batch 64KB score tile DMA'd by the Tensor Data Mover,
//     overlapped with the sigmoid precompute pass.
// LDS: sA 64KB (scores -> inside chart, in place) + sG 64KB (grad) + sMid 8KB.

#define CKY_L    128
#define CKY_LL   (CKY_L * CKY_L)
#define CKY_T    16
#define CKY_NBLK (CKY_L / CKY_T)   // 8 == waves per block

typedef unsigned int cky_u32;
typedef __attribute__((ext_vector_type(4))) cky_u32 cky_u32x4;
typedef __attribute__((ext_vector_type(8))) cky_u32 cky_u32x8;
typedef __attribute__((ext_vector_type(2))) float cky_v2f;
typedef __attribute__((ext_vector_type(8))) float cky_v8f;

#if defined(__has_builtin)
# if __has_builtin(__builtin_amdgcn_wmma_f32_16x16x4_f32)
#  define CKY_HAVE_WMMA 1
# endif
#endif
#ifndef CKY_HAVE_WMMA
# define CKY_HAVE_WMMA 0
#endif

__device__ __forceinline__ float cky_log1p_exp(float s) {
    return fmaxf(s, 0.0f) + log1pf(__expf(-fabsf(s)));
}

// online logsumexp merge into (mrun, accr)
#define CKY_MERGE(vexpr) do {                                   \
    float v_ = (vexpr);                                         \
    if (v_ <= mrun) { accr += __expf(v_ - mrun); }              \
    else { accr = accr * __expf(mrun - v_) + 1.0f; mrun = v_; } \
} while (0)

__global__ __launch_bounds__(256) void cky_inside_outside_kernel(
    const float* __restrict__ scores,   // (B, L, L, 1) fp32
    const int*   __restrict__ seq_lens, // (B,) int32
    float*       __restrict__ out_Z,    // (B,)
    float*       __restrict__ out_marg) // (B, L, L)
{
    __shared__ __align__(16) float sA[CKY_LL];                     // scores -> inside chart
    __shared__ __align__(16) float sG[CKY_LL];                     // dZ/dA
    __shared__ __align__(16) float sMid[CKY_NBLK * CKY_T * CKY_T]; // per-wave middle-LSE tile

    const int b    = blockIdx.x;
    const int t    = threadIdx.x;
    const int NT   = blockDim.x;
    const int lane = t & 31;
    const int wv   = t >> 5;       // wave id, 0..7
    const int hi   = lane >> 4;    // which 16-lane half
    const int ln   = lane & 15;

    const float* sb = scores + (size_t)b * CKY_LL;
    float*       mb = out_marg + (size_t)b * CKY_LL;

    // ---------------------------------------------------------------
    // TDM: issue DMA of the 128x128 fp32 score tile into LDS (wave 0).
    // D# group0/group1 per CDNA5 ISA sec. 8.3/8.4.
    // ---------------------------------------------------------------
    if (t < 32) {
        unsigned long long ga = (unsigned long long)(const void*)sb;
        cky_u32 lds_off = (cky_u32)(unsigned long long)(void*)&sA[0];

        cky_u32x4 g0;
        g0.x = 1u;                                       // count=1
        g0.y = lds_off;
        g0.z = (cky_u32)(ga & 0xffffffffull);
        g0.w = (cky_u32)((ga >> 32) & 0x01ffffffull) | (2u << 30); // type=2

        cky_u32x8 g1;
        g1.s0 = (2u << 16);                 // data_size = 4B
        g1.s1 = ((cky_u32)CKY_L) << 16;     // tensor_dim0 = 128
        g1.s2 = ((cky_u32)CKY_L) << 16;     // tensor_dim1 = 128
        g1.s3 = ((cky_u32)CKY_L) << 16;     // tile_dim0 = 128
        g1.s4 = (cky_u32)CKY_L;             // tile_dim1 = 128
        g1.s5 = (cky_u32)CKY_L;             // tensor_dim0_stride = 128
        g1.s6 = 0u;
        g1.s7 = 0u;

        asm volatile("tensor_load_to_lds %0, %1" :: "s"(g0), "s"(g1) : "memory");
    }

    // ---------------------------------------------------------------
    // Phase 0 (overlaps the DMA): t[i,j] = sigmoid(-s) = e^{-c_ij} into mb.
    // Backward uses it as the exact softmax normalizer; final marginal is
    // g * (1 - t). Writing every index also makes the output deterministic.
    // ---------------------------------------------------------------
    for (int idx = t; idx < CKY_LL; idx += NT) {
        mb[idx] = 1.0f / (1.0f + __expf(sb[idx]));
    }

    if (t < 32) __builtin_amdgcn_s_wait_tensorcnt(0);
    __syncthreads();

    // ---------------------------------------------------------------
    // Forward, block diagonal d = 0: in-block CKY (one 16x16 diag block/wave)
    // ---------------------------------------------------------------
    {
        const int r0 = CKY_T * wv;
        if (hi == 0) {
            const int i = r0 + ln;
            sA[i * CKY_L + i] = cky_log1p_exp(sA[i * CKY_L + i]);
        }
        asm volatile("" ::: "memory");
        for (int w = 1; w <= CKY_T - 1; ++w) {
            const int i = r0 + ln;
            const int j = i + w;
            if (hi == 0 && ln + w <= CKY_T - 1) {
                float mrun = -3.4e38f, accr = 0.0f;
                for (int k = i; k < j; ++k)
                    CKY_MERGE(sA[i * CKY_L + k] + sA[(k + 1) * CKY_L + j]);
                float lse = mrun + __logf(accr);
                sA[i * CKY_L + j] = lse + cky_log1p_exp(sA[i * CKY_L + j]);
            }
            asm volatile("" ::: "memory");
        }
    }
    __syncthreads();

    // ---------------------------------------------------------------
    // Forward, block diagonals d = 1..7 (block pair (p, p+d) per wave).
    // ---------------------------------------------------------------
    for (int d = 1; d < CKY_NBLK; ++d) {
        const int p = wv;
        if (p < CKY_NBLK - d) {
            const int q  = p + d;
            const int r0 = CKY_T * p;
            const int c0 = CKY_T * q;
            float* mid = &sMid[wv * CKY_T * CKY_T];

#if CKY_HAVE_WMMA
            if (d >= 2) {
                // ---- middle split blocks m in (p,q): log-matmul-exp via WMMA ----
                float Mx[8], Sx[8];
#pragma unroll
                for (int r = 0; r < 8; ++r) { Mx[r] = -3.4e38f; Sx[r] = 0.0f; }

                for (int m = p + 1; m < q; ++m) {
                    const int k0 = CKY_T * m;
                    const float* lrow = &sA[(r0 + ln) * CKY_L + k0]; // L-tile row ln
                    const float* rcol = &sA[(k0 + 1) * CKY_L + c0 + ln]; // R-tile col ln (shifted)

                    // row max of L-tile row ln, col max of R-tile col ln
                    float rmax = lrow[0], cmax = rcol[0];
#pragma unroll
                    for (int kk = 1; kk < CKY_T; ++kk) {
                        rmax = fmaxf(rmax, lrow[kk]);
                        cmax = fmaxf(cmax, rcol[kk * CKY_L]);
                    }
                    // P[M,N] = sum_k exp(L-rmax)*exp(R-cmax), K=16 via 4 WMMA chunks.
                    // ISA 16x4 f32 operand layout: lanes 0-15 supply K=4c,4c+1;
                    // lanes 16-31 supply K=4c+2,4c+3. The lane-divergent element
                    // select is done in the LDS *address* (native ds_load), never
                    // as a register-indexed local array.
                    cky_v8f acc = {};
#pragma unroll
                    for (int c = 0; c < 4; ++c) {
                        const int ka = 4 * c + 2 * hi;
                        cky_v2f av, bv;
                        av.x = __expf(lrow[ka]                - rmax);
                        av.y = __expf(lrow[ka + 1]            - rmax);
                        bv.x = __expf(rcol[ka * CKY_L]        - cmax);
                        bv.y = __expf(rcol[(ka + 1) * CKY_L]  - cmax);
                        acc = __builtin_amdgcn_wmma_f32_16x16x4_f32(
                                  false, av, false, bv, (short)0, acc, false, false);
                    }
                    // Ct[M,N] = rmax[M] + cmax[N] + log(P); rmax[M] via ds_bpermute
                    // (lane M holds rmax of row M; byte address = lane*4).
#pragma unroll
                    for (int r = 0; r < 8; ++r) {
                        const int Mrow = hi * 8 + r;
                        float rmaxM = __int_as_float(__builtin_amdgcn_ds_bpermute(
                                          Mrow << 2, __float_as_int(rmax)));
                        float ct = rmaxM + cmax + __logf(acc[r]);
                        if (ct <= Mx[r]) { Sx[r] += __expf(ct - Mx[r]); }
                        else { Sx[r] = Sx[r] * __expf(Mx[r] - ct) + 1.0f; Mx[r] = ct; }
                    }
                }
                // spill middle-LSE per cell to per-wave LDS scratch
#pragma unroll
                for (int r = 0; r < 8; ++r) {
                    float cm = (Sx[r] > 0.0f) ? (Mx[r] + __logf(Sx[r])) : -3.0e38f;
                    mid[(hi * 8 + r) * CKY_T + ln] = cm;
                }
            }
#endif
            asm volatile("" ::: "memory");

            // ---- boundary sweep: widths 16d-15 .. 16d+15, cell (r0+ln, +w) ----
            const int wmin = CKY_T * d - (CKY_T - 1);
            for (int s = 0; s <= 2 * (CKY_T - 1); ++s) {
                const int w = wmin + s;
                const int i = r0 + ln;
                const int j = i + w;
                if (hi == 0 && j >= c0 && j <= c0 + CKY_T - 1) {
                    float mrun, accr;
#if CKY_HAVE_WMMA
                    if (d >= 2) {
                        float cm = mid[ln * CKY_T + (j - c0)];
                        if (cm > -2.9e38f) { mrun = cm; accr = 1.0f; }
                        else               { mrun = -3.4e38f; accr = 0.0f; }
                    } else
#endif
                    { mrun = -3.4e38f; accr = 0.0f; }

                    // splits with k in block p (right child internal or block p+1)
                    for (int k = i; k <= r0 + CKY_T - 1; ++k)
                        CKY_MERGE(sA[i * CKY_L + k] + sA[(k + 1) * CKY_L + j]);
#if !CKY_HAVE_WMMA
                    // scalar fallback: interior split blocks
                    for (int k = r0 + CKY_T; k <= c0 - 1; ++k)
                        CKY_MERGE(sA[i * CKY_L + k] + sA[(k + 1) * CKY_L + j]);
#endif
                    // splits with k in block q (left child internal, right in (q,q))
                    for (int k = c0; k < j; ++k)
                        CKY_MERGE(sA[i * CKY_L + k] + sA[(k + 1) * CKY_L + j]);

                    float lse = mrun + __logf(accr);
                    sA[i * CKY_L + j] = lse + cky_log1p_exp(sA[i * CKY_L + j]);
                }
                asm volatile("" ::: "memory");
            }
        }
        __syncthreads();
    }

    int len = seq_lens[b];
    if (len < 1) len = 1;
    if (len > CKY_L) len = CKY_L;

    if (t == 0) out_Z[b] = sA[len - 1];       // A[0, len-1]

    // ---------------------------------------------------------------
    // Backward: g = dZ/dA. Split posterior p_k = exp(A[i,k]+A[k+1,j]-A[i,j])/Q
    // with Q = e^{-c_ij} = t[i,j] taken analytically from the phase-0 pass.
    // ---------------------------------------------------------------
    for (int idx = t; idx < CKY_LL; idx += NT) sG[idx] = 0.0f;
    __syncthreads();
    if (t == 0) sG[len - 1] = 1.0f;
    __syncthreads();

    for (int w = len - 1; w >= 1; --w) {
        for (int i = t; i <= len - 1 - w; i += NT) {
            const int j = i + w;
            const float g = sG[i * CKY_L + j];
            if (g != 0.0f) {
                const float a    = sA[i * CKY_L + j];
                const float tq   = fmaxf(mb[i * CKY_L + j], 1e-38f); // Q = e^{-c}
                const float ginv = g / tq;
                for (int k = i; k < j; ++k) {
                    float pp = __expf(sA[i * CKY_L + k] + sA[(k + 1) * CKY_L + j] - a) * ginv;
                    atomicAdd(&sG[i * CKY_L + k], pp);
                    atomicAdd(&sG[(k + 1) * CKY_L + j], pp);
                }
            }
        }
        __syncthreads();
    }

    // marginal[i,j] = g[i,j] * sigmoid(s[i,j]) = g * (1 - t)
    for (int idx = t; idx < CKY_LL; idx += NT) {
        mb[idx] = sG[idx] * (1.0f - mb[idx]);
    }
}

extern "C" void kernel_launch(void* const* d_in, const int* in_sizes, int n_in,
                              void* d_out, int out_size, void* d_ws, size_t ws_size,
                              hipStream_t stream) {
    const float* scores = (const float*)d_in[0];   // (B, L, L, 1) fp32
    const int*   lens   = (const int*)d_in[1];     // (B,) int32
    const int B = in_sizes[1];                     // 32

    float* outZ = (float*)d_out;                   // (B,)
    float* outM = outZ + B;                        // (B, L, L)

    cky_inside_outside_kernel<<<B, 256, 0, stream>>>(scores, lens, outZ, outM);
}